// MViTV2SelfAttentionPoolingFirst_51402168599291
// MI455X (gfx1250) — compile-verified
//
#include <hip/hip_runtime.h>
#include <math.h>

typedef __attribute__((ext_vector_type(2))) float v2f;
typedef __attribute__((ext_vector_type(4))) float v4f;
typedef __attribute__((ext_vector_type(8))) float v8f;

#define IN_DIM   96
#define OUT_DIM  192
#define NHEADS   2
#define HDIM     96
#define HFS      56
#define WFS      56
#define BATCH    64
#define NSRC     (HFS*WFS + 1)      // 3137
#define QHH      28
#define QWW      28
#define KHH      14
#define KWW      14
#define NQ       (QHH*QWW + 1)      // 785
#define NK       (KHH*KWW + 1)      // 197
#define NKP      208                // 13 * 16 padded key dim
#define NQ_TILES 50                 // ceil(785/16)
#define NK_TILES 13
#define KSTEPS   (HDIM/4)           // 24 WMMA k-steps over head dim
#define LN_EPS   1e-5f

// ---------------------------------------------------------------------------
// Kernel 1: depthwise 3x3 conv (pad 1) + cls passthrough + LayerNorm over C=96
// grid = (ntok, B), block = 96 threads (one per channel)
// ---------------------------------------------------------------------------
__global__ __launch_bounds__(96)
void pool_ln_kernel(const float* __restrict__ hs, const float* __restrict__ w,
                    const float* __restrict__ g,  const float* __restrict__ beta,
                    float* __restrict__ out, int stride, int oh, int ow) {
  const int tok = blockIdx.x;
  const int b   = blockIdx.y;
  const int c   = threadIdx.x;
  const int ntok = oh * ow + 1;

  float val;
  if (tok == 0) {
    val = hs[(size_t)(b * NSRC) * IN_DIM + c];
  } else {
    const int sp = tok - 1;
    const int oy = sp / ow, ox = sp % ow;
    float acc = 0.f;
#pragma unroll
    for (int dy = 0; dy < 3; ++dy) {
      const int iy = oy * stride - 1 + dy;
      if (iy < 0 || iy >= HFS) continue;
#pragma unroll
      for (int dx = 0; dx < 3; ++dx) {
        const int ix = ox * stride - 1 + dx;
        if (ix < 0 || ix >= WFS) continue;
        acc += hs[(size_t)(b * NSRC + 1 + iy * WFS + ix) * IN_DIM + c] * w[c * 9 + dy * 3 + dx];
      }
    }
    val = acc;
  }

  __shared__ float red[96];
  __shared__ float s_mean, s_rstd;
  red[c] = val;
  __syncthreads();
  if (c < 32) {
    float s = red[c] + red[c + 32] + red[c + 64];
    for (int o = 16; o > 0; o >>= 1) s += __shfl_down(s, o);
    if (c == 0) s_mean = s * (1.f / 96.f);
  }
  __syncthreads();
  const float d = val - s_mean;
  red[c] = d * d;
  __syncthreads();
  if (c < 32) {
    float s = red[c] + red[c + 32] + red[c + 64];
    for (int o = 16; o > 0; o >>= 1) s += __shfl_down(s, o);
    if (c == 0) s_rstd = rsqrtf(s * (1.f / 96.f) + LN_EPS);
  }
  __syncthreads();
  out[(size_t)(b * ntok + tok) * IN_DIM + c] = (val - s_mean) * s_rstd * g[c] + beta[c];
}

// ---------------------------------------------------------------------------
// Kernel 2: projection GEMM X(M x 96) @ W(96 x 192) + bias -> [B][H][n][96]
// one wave per 16x16 tile; grid = (M/16, 192/16), block = 32
// ---------------------------------------------------------------------------
__global__ __launch_bounds__(32)
void proj_kernel(const float* __restrict__ X, const float* __restrict__ W,
                 const float* __restrict__ bias, float* __restrict__ out,
                 int M, int ntok) {
  const int l  = threadIdx.x;
  const int hl = l >> 4;          // half: 0 or 1
  const int lm = l & 15;
  const int mbase = blockIdx.x * 16;
  const int nbase = blockIdx.y * 16;

  int row = mbase + lm; if (row >= M) row = M - 1;
  const float* __restrict__ xr = X + (size_t)row * IN_DIM;
  const float* __restrict__ wc = W + nbase + lm;

  v8f acc = {};
  for (int k = 0; k < IN_DIM; k += 4) {
    const int klo = k + 2 * hl;
    v2f a = *(const v2f*)(xr + klo);
    v2f bm;
    bm.x = wc[(size_t)klo * OUT_DIM];
    bm.y = wc[(size_t)(klo + 1) * OUT_DIM];
    acc = __builtin_amdgcn_wmma_f32_16x16x4_f32(false, a, false, bm, (short)0, acc,
                                                false, false);
  }

  const int col = nbase + lm;              // 0..191
  const float bc = bias[col];
  const int hd = col / HDIM;               // head (uniform per tile)
  const int dd = col % HDIM;
#pragma unroll
  for (int i = 0; i < 8; ++i) {
    const int m = mbase + i + 8 * hl;
    if (m < M) {
      const int bb = m / ntok, t = m % ntok;
      out[(size_t)((bb * NHEADS + hd) * ntok + t) * HDIM + dd] = acc[i] + bc;
    }
  }
}

// ---------------------------------------------------------------------------
// Kernel 3: attention. one wave per (batch, head, 16-query tile)
// grid = (NQ_TILES, NHEADS, BATCH), block = 32
// ---------------------------------------------------------------------------
__global__ __launch_bounds__(32)
void attn_kernel(const float* __restrict__ Qp, const float* __restrict__ Kp,
                 const float* __restrict__ Vp, const float* __restrict__ relh,
                 const float* __restrict__ relw, float* __restrict__ out) {
  const int l  = threadIdx.x;
  const int hl = l >> 4;
  const int lm = l & 15;
  const int qt = blockIdx.x;
  const int hd = blockIdx.y;
  const int b  = blockIdx.z;
  const int qbase = qt * 16;
  const float scale = rsqrtf((float)HDIM);

  const float* __restrict__ Q = Qp + (size_t)((b * NHEADS + hd) * NQ) * HDIM;
  const float* __restrict__ K = Kp + (size_t)((b * NHEADS + hd) * NK) * HDIM;
  const float* __restrict__ V = Vp + (size_t)((b * NHEADS + hd) * NK) * HDIM;

  __shared__ float s_eh[16][14];
  __shared__ float s_ew[16][14];
  __shared__ float s_S[16][NKP];
  __shared__ float s_sum[16];

  // ---- 1) rel-pos bias dot products (decomposed index: i - 2j + 26) ----
  // lanes 0..13 compute eh[*, lm]; lanes 16..29 compute ew[*, lm]
  if (lm < 14) {
    for (int m = 0; m < 16; ++m) {
      const int t = qbase + m;
      float acc = 0.f;
      if (t >= 1 && t < NQ) {
        const int sp = t - 1;
        const int qy = sp / QWW, qx = sp % QWW;
        const int ridx = (hl == 0) ? (qy - 2 * lm + 26) : (qx - 2 * lm + 26);
        const float* __restrict__ R = ((hl == 0) ? relh : relw) + (size_t)ridx * HDIM;
        const float* __restrict__ q = Q + (size_t)t * HDIM;
        for (int c = 0; c < HDIM; c += 4) {
          v4f qv = *(const v4f*)(q + c);
          v4f rv = *(const v4f*)(R + c);
          acc += qv.x * rv.x + qv.y * rv.y + qv.z * rv.z + qv.w * rv.w;
        }
      }
      if (hl == 0) s_eh[m][lm] = acc; else s_ew[m][lm] = acc;
    }
  }

  // ---- 2) S = scale * Q K^T via WMMA f32 16x16x4 ----
  // Hoist the A operand (this wave's Q tile share: 24 x v2f = 48 regs) once.
  int qrow = qbase + lm; if (qrow >= NQ) qrow = NQ - 1;
  const float* __restrict__ qr = Q + (size_t)qrow * HDIM;
  v2f aq[KSTEPS];
#pragma unroll
  for (int i = 0; i < KSTEPS; ++i) aq[i] = *(const v2f*)(qr + 4 * i + 2 * hl);

  for (int j = 0; j < NK_TILES; ++j) {
    int krow = j * 16 + lm; if (krow >= NK) krow = NK - 1;
    const float* __restrict__ kr = K + (size_t)krow * HDIM;
    v8f acc = {};
#pragma unroll
    for (int i = 0; i < KSTEPS; ++i) {
      v2f bm = *(const v2f*)(kr + 4 * i + 2 * hl);
      acc = __builtin_amdgcn_wmma_f32_16x16x4_f32(false, aq[i], false, bm, (short)0,
                                                  acc, false, false);
    }
#pragma unroll
    for (int i = 0; i < 8; ++i) s_S[i + 8 * hl][j * 16 + lm] = acc[i] * scale;
  }
  __syncthreads();

  // ---- 3) bias add + masked softmax (2 lanes per row, strided cols) ----
  {
    const int r = lm;
    const int t = qbase + r;
    float mx = -1e30f;
    for (int c = hl; c < NK; c += 2) {
      float s = s_S[r][c];
      if (t >= 1 && t < NQ && c >= 1) {
        const int ky = (c - 1) / KWW, kx = (c - 1) % KWW;
        s += s_eh[r][ky] + s_ew[r][kx];
      }
      s_S[r][c] = s;
      mx = fmaxf(mx, s);
    }
    mx = fmaxf(mx, __shfl_xor(mx, 16));
    float sum = 0.f;
    for (int c = hl; c < NKP; c += 2) {
      const float p = (c < NK) ? __expf(s_S[r][c] - mx) : 0.f;
      s_S[r][c] = p;
      sum += p;
    }
    sum += __shfl_xor(sum, 16);
    if (hl == 0) s_sum[r] = sum;
  }
  __syncthreads();

  // ---- 4) O = P V via WMMA: outer K, inner 6 dim tiles (A loaded once/K) ----
  v8f oacc[6];
#pragma unroll
  for (int j = 0; j < 6; ++j) oacc[j] = (v8f){};

  for (int k = 0; k < NKP; k += 4) {
    const int klo = k + 2 * hl;
    v2f a = *(const v2f*)(&s_S[lm][klo]);
    int vr0 = klo;     if (vr0 >= NK) vr0 = NK - 1;
    int vr1 = klo + 1; if (vr1 >= NK) vr1 = NK - 1;
    const float* __restrict__ v0 = V + (size_t)vr0 * HDIM + lm;
    const float* __restrict__ v1 = V + (size_t)vr1 * HDIM + lm;
#pragma unroll
    for (int j = 0; j < 6; ++j) {
      v2f bm;
      bm.x = v0[j * 16];
      bm.y = v1[j * 16];
      oacc[j] = __builtin_amdgcn_wmma_f32_16x16x4_f32(false, a, false, bm, (short)0,
                                                      oacc[j], false, false);
    }
  }

  // ---- 5) /rowsum, +Q residual, store ----
#pragma unroll
  for (int j = 0; j < 6; ++j) {
    const int d = j * 16 + lm;
#pragma unroll
    for (int i = 0; i < 8; ++i) {
      const int m = i + 8 * hl;
      const int t = qbase + m;
      if (t < NQ) {
        float val = oacc[j][i] / s_sum[m];
        if (t >= 1) val += Q[(size_t)t * HDIM + d];
        out[(size_t)(b * NQ + t) * OUT_DIM + hd * HDIM + d] = val;
      }
    }
  }
}

// ---------------------------------------------------------------------------
// Host launcher
// ---------------------------------------------------------------------------
extern "C" void kernel_launch(void* const* d_in, const int* in_sizes, int n_in,
                              void* d_out, int out_size, void* d_ws, size_t ws_size,
                              hipStream_t stream) {
  (void)in_sizes; (void)n_in; (void)out_size; (void)ws_size;
  const float* hs   = (const float*)d_in[0];
  const float* Wq   = (const float*)d_in[1];
  const float* bq   = (const float*)d_in[2];
  const float* Wk   = (const float*)d_in[3];
  const float* bk   = (const float*)d_in[4];
  const float* Wv   = (const float*)d_in[5];
  const float* bv   = (const float*)d_in[6];
  const float* pqw  = (const float*)d_in[7];
  const float* pkw  = (const float*)d_in[8];
  const float* pvw  = (const float*)d_in[9];
  const float* gq   = (const float*)d_in[10];
  const float* betq = (const float*)d_in[11];
  const float* gk   = (const float*)d_in[12];
  const float* betk = (const float*)d_in[13];
  const float* gv   = (const float*)d_in[14];
  const float* betv = (const float*)d_in[15];
  const float* relh = (const float*)d_in[16];
  const float* relw = (const float*)d_in[17];
  float* outp = (float*)d_out;

  float* qpool = (float*)d_ws;                       // B*NQ*96
  float* kpool = qpool + (size_t)BATCH * NQ * IN_DIM;
  float* vpool = kpool + (size_t)BATCH * NK * IN_DIM;
  float* qproj = vpool + (size_t)BATCH * NK * IN_DIM;        // B*H*NQ*96
  float* kproj = qproj + (size_t)BATCH * NHEADS * NQ * HDIM;
  float* vproj = kproj + (size_t)BATCH * NHEADS * NK * HDIM;

  // 1) pooling + LN
  pool_ln_kernel<<<dim3(NQ, BATCH), 96, 0, stream>>>(hs, pqw, gq, betq, qpool, 2, QHH, QWW);
  pool_ln_kernel<<<dim3(NK, BATCH), 96, 0, stream>>>(hs, pkw, gk, betk, kpool, 4, KHH, KWW);
  pool_ln_kernel<<<dim3(NK, BATCH), 96, 0, stream>>>(hs, pvw, gv, betv, vpool, 4, KHH, KWW);

  // 2) projections (M divisible by 16: 50240 and 12608)
  const int Mq = BATCH * NQ, Mk = BATCH * NK;
  proj_kernel<<<dim3(Mq / 16, OUT_DIM / 16), 32, 0, stream>>>(qpool, Wq, bq, qproj, Mq, NQ);
  proj_kernel<<<dim3(Mk / 16, OUT_DIM / 16), 32, 0, stream>>>(kpool, Wk, bk, kproj, Mk, NK);
  proj_kernel<<<dim3(Mk / 16, OUT_DIM / 16), 32, 0, stream>>>(vpool, Wv, bv, vproj, Mk, NK);

  // 3) attention + rel-pos + softmax + PV + residual
  attn_kernel<<<dim3(NQ_TILES, NHEADS, BATCH), 32, 0, stream>>>(qproj, kproj, vproj,
                                                                relh, relw, outp);
}